// YOLOLossv3_52097953300961
// MI455X (gfx1250) — compile-verified
//
#include <hip/hip_runtime.h>

typedef __attribute__((ext_vector_type(2))) float v2f;
typedef __attribute__((ext_vector_type(8))) float v8f;

#define NB 32
#define NA 3
#define NH 128
#define NW 128
#define NCHAN 18            // NA * (5 + CLS_NUM)
#define CH_STRIDE (NH * NW) // 16384
#define NCELL (NB * NA * NH * NW) // 1,572,864
#define BLK 256
#define NBLK 1536           // NBLK*BLK*4 == NCELL ; NBLK % 64 == 0
#define EPS 1e-7f
#define OBJ_BIT (1u << 30)

__constant__ float ANCH_W[3] = {116.f, 156.f, 373.f};
__constant__ float ANCH_H[3] = {90.f, 198.f, 326.f};

__device__ __forceinline__ float sigmoidf_(float x) {
    return 1.0f / (1.0f + __expf(-x));
}

// Deterministic wave-level sum of n floats (n % 64 == 0) using the matrix pipe:
// D = A(16x4, data) * B(4x16, ones) + C  ->  D[m][n] = rowsum_m(A) + C[m][n].
// Any placement of the 64 chunk values into A's 2 VGPRs is valid since B is all
// ones. Column 0 of D holds the 16 row sums: m=0..7 live in VGPR m lane 0,
// m=8..15 in VGPR m-8 lane 16 -> total = shfl(sum_r d[r], 0) + shfl(..., 16).
// Requires all 32 lanes of the wave active (EXEC == ~0 for WMMA).
__device__ __forceinline__ float wave_wmma_sum(const float* buf, int n, int lane) {
    v8f acc = {};
    v2f ones;
    ones.x = 1.0f;
    ones.y = 1.0f;
    for (int c = 0; c < n; c += 64) {
        v2f a;
        a.x = buf[c + lane];
        a.y = buf[c + 32 + lane];
        acc = __builtin_amdgcn_wmma_f32_16x16x4_f32(
            false, a, false, ones, (short)0, acc, false, false);
    }
    float v = acc[0] + acc[1] + acc[2] + acc[3] + acc[4] + acc[5] + acc[6] + acc[7];
    return __shfl(v, 0, 32) + __shfl(v, 16, 32);
}

// ---- Kernel 1: build targets (scatter). One thread per ground-truth box. ----
// mask[cell] bit30  : "noobj cleared" (cell is best-anchor target or iou>thres)
// mask[cell] [15:0] : winner GT index + 1 (0 = no obj target here)
// atomicOr / atomicMax commute -> deterministic fixpoint regardless of order.
__global__ void yolo_scatter(const int* __restrict__ gt_batch,
                             const float* __restrict__ gt_boxes, int nG,
                             const int* __restrict__ size_h,
                             const int* __restrict__ size_w,
                             unsigned* __restrict__ mask) {
    int g = blockIdx.x * blockDim.x + threadIdx.x;
    if (g >= nG) return;
    float stw = (float)((*size_w) / NW); // stride_w
    float sth = (float)((*size_h) / NH); // stride_h

    int b = gt_batch[g];
    float4 box = ((const float4*)gt_boxes)[g];
    float gx = box.x * NW, gy = box.y * NH;
    float gw = box.z * NW, gh = box.w * NH;
    int gi = (int)gx; gi = gi < 0 ? 0 : (gi > NW - 1 ? NW - 1 : gi);
    int gj = (int)gy; gj = gj < 0 ? 0 : (gj > NH - 1 ? NH - 1 : gj);

    float best_iou = -1.0f;
    int best = 0;
    for (int a = 0; a < NA; ++a) {
        float saw = ANCH_W[a] / stw;
        float sah = ANCH_H[a] / sth;
        float inter = fminf(gw, saw) * fminf(gh, sah);
        float uni = gw * gh + saw * sah - inter;
        float iou = inter / uni;
        if (iou > best_iou) { best_iou = iou; best = a; } // strict > == argmax-first
        if (iou > 0.5f)
            atomicOr(&mask[(((b * NA) + a) << 14) + gj * NW + gi], OBJ_BIT);
    }
    atomicMax(&mask[(((b * NA) + best) << 14) + gj * NW + gi],
              OBJ_BIT | (unsigned)(g + 1));
}

// ---- Kernel 2: main masked reduction. 4 cells/thread, float4/uint4 loads. ----
// Only the conf channel is streamed for every cell; x/y/w/h channels and the
// GT box are fetched only on the rare obj path. Block partials reduced
// deterministically by wave 0 via WMMA row-sum trick.
__global__ void __launch_bounds__(BLK) yolo_main(
    const float* __restrict__ out, const float* __restrict__ gt_boxes,
    const int* __restrict__ size_h, const int* __restrict__ size_w,
    const unsigned* __restrict__ mask, float* __restrict__ partial) {
    __shared__ float red[4][BLK];

    int t = blockIdx.x * BLK + threadIdx.x;
    int n0 = t * 4;                 // first of 4 consecutive cells
    int ba = n0 >> 14;              // b*NA + a   (cells within one (b,a) plane: 16384, %4==0)
    int b = ba / NA;
    int a = ba - b * NA;
    int rem = n0 & (CH_STRIDE - 1);
    int base = ((b * NCHAN + a * 6) << 14) + rem;

    float4 cv = *(const float4*)(out + base + 4 * CH_STRIDE); // conf logits
    uint4 mv = *(const uint4*)(mask + n0);

    float stw = (float)((*size_w) / NW);
    float sth = (float)((*size_h) / NH);
    float inv_saw = stw / ANCH_W[a]; // 1 / scaled_anchor_w
    float inv_sah = sth / ANCH_H[a];

    float cvv[4] = {cv.x, cv.y, cv.z, cv.w};
    unsigned mvv[4] = {mv.x, mv.y, mv.z, mv.w};

    float s_obj = 0.f, s_noobj = 0.f, c_obj = 0.f, c_noobj = 0.f;
    for (int k = 0; k < 4; ++k) {
        float p = sigmoidf_(cvv[k]);
        p = fminf(fmaxf(p, EPS), 1.0f - EPS);
        unsigned m = mvv[k];
        if (!(m & OBJ_BIT)) { // noobj cell: BCE with tconf=0
            s_noobj -= __logf(1.0f - p);
            c_noobj += 1.0f;
        } else if (m & 0xFFFFu) { // obj cell (winner present)
            int w = (int)(m & 0xFFFFu) - 1;
            float4 box = ((const float4*)gt_boxes)[w];
            float gx = box.x * NW, gy = box.y * NH;
            float gw = box.z * NW, gh = box.w * NH;
            float tx = gx - floorf(gx);
            float ty = gy - floorf(gy);
            float tw = __logf(gw * inv_saw);
            float th = __logf(gh * inv_sah);
            int cb = base + k;
            float xs = sigmoidf_(out[cb]);
            float ys = sigmoidf_(out[cb + CH_STRIDE]);
            float ws = out[cb + 2 * CH_STRIDE];
            float hs = out[cb + 3 * CH_STRIDE];
            float dx = xs - tx, dy = ys - ty, dw = ws - tw, dh = hs - th;
            s_obj += dx * dx + dy * dy + dw * dw + dh * dh - __logf(p);
            c_obj += 1.0f;
        }
        // cells cleared by ignore-thres but not a best-anchor target: no term
    }

    red[0][threadIdx.x] = s_obj;
    red[1][threadIdx.x] = s_noobj;
    red[2][threadIdx.x] = c_obj;
    red[3][threadIdx.x] = c_noobj;
    __syncthreads();

    if (threadIdx.x < 32) { // wave 0, all 32 lanes active -> WMMA legal
        for (int c = 0; c < 4; ++c) {
            float tot = wave_wmma_sum(&red[c][0], BLK, threadIdx.x);
            if (threadIdx.x == 0) partial[c * NBLK + blockIdx.x] = tot;
        }
    }
}

// ---- Kernel 3: final 1-wave WMMA reduction of block partials + loss. ----
__global__ void yolo_final(const float* __restrict__ partial,
                           float* __restrict__ loss) {
    int lane = threadIdx.x;
    float tot0 = wave_wmma_sum(partial + 0 * NBLK, NBLK, lane);
    float tot1 = wave_wmma_sum(partial + 1 * NBLK, NBLK, lane);
    float tot2 = wave_wmma_sum(partial + 2 * NBLK, NBLK, lane);
    float tot3 = wave_wmma_sum(partial + 3 * NBLK, NBLK, lane);
    if (lane == 0) {
        float n_obj = fmaxf(tot2, 1.0f);
        float n_noobj = fmaxf(tot3, 1.0f);
        // loss = (bbox + obj-BCE)/n_obj + NOOBJ_SCALE * noobj-BCE / n_noobj
        loss[0] = tot0 / n_obj + 100.0f * tot1 / n_noobj;
    }
}

extern "C" void kernel_launch(void* const* d_in, const int* in_sizes, int n_in,
                              void* d_out, int out_size, void* d_ws, size_t ws_size,
                              hipStream_t stream) {
    (void)n_in; (void)out_size; (void)ws_size;
    const float* out = (const float*)d_in[0];
    const int* gt_batch = (const int*)d_in[1];
    const float* gt_boxes = (const float*)d_in[2];
    const int* size_h = (const int*)d_in[3];
    const int* size_w = (const int*)d_in[4];
    int nG = in_sizes[1];

    unsigned* mask = (unsigned*)d_ws;                                   // NCELL u32 (6.29 MB)
    float* partial = (float*)((char*)d_ws + (size_t)NCELL * sizeof(unsigned)); // 4*NBLK f32

    hipMemsetAsync(mask, 0, (size_t)NCELL * sizeof(unsigned), stream);
    yolo_scatter<<<(nG + 255) / 256, 256, 0, stream>>>(gt_batch, gt_boxes, nG,
                                                       size_h, size_w, mask);
    yolo_main<<<NBLK, BLK, 0, stream>>>(out, gt_boxes, size_h, size_w, mask, partial);
    yolo_final<<<1, 32, 0, stream>>>(partial, (float*)d_out);
}